// Sample_Group_24300924961392
// MI455X (gfx1250) — compile-verified
//
#include <hip/hip_runtime.h>
#include <hip/hip_bf16.h>

typedef __attribute__((ext_vector_type(2))) float v2f;
typedef __attribute__((ext_vector_type(8))) float v8f;
typedef unsigned u32x4 __attribute__((ext_vector_type(4)));
typedef int      i32x4 __attribute__((ext_vector_type(4)));
typedef int      i32x8 __attribute__((ext_vector_type(8)));

#define BATCH   8
#define NPTS    4096
#define NPOINT  1024
#define KNBR    32
#define CH      64

// Generic pointers into LDS carry the LDS byte offset in their low 32 bits
// (LDS aperture rule).
__device__ __forceinline__ unsigned lds_off_u32(const void* p) {
    return (unsigned)(unsigned long long)p;
}

__device__ __forceinline__ void wait_asynccnt0() {
#if __has_builtin(__builtin_amdgcn_s_wait_asynccnt)
    __builtin_amdgcn_s_wait_asynccnt(0);
#else
    asm volatile("s_wait_asynccnt 0x0" ::: "memory");
#endif
}

// ---------------------------------------------------------------------------
// Kernel 1: farthest point sampling. One workgroup per batch.
// Batch xyz (48KB AoS) staged into LDS with async global->LDS b128 copies;
// running min-dist also LDS-resident. 1024 sequential steps of
// (dist-to-centroid, min-update, argmax), argmax ties -> lowest index.
// ---------------------------------------------------------------------------
__global__ __launch_bounds__(256) void fps_kernel(const float* __restrict__ xyz,
                                                  float* __restrict__ new_xyz,
                                                  int* __restrict__ fps_idx) {
    extern __shared__ float sm[];
    float* lxyz = sm;                 // 12288 floats, AoS (x,y,z)
    float* dist = sm + 3 * NPTS;      // 4096
    __shared__ float rval[8];
    __shared__ int   ridx[8];
    __shared__ int   s_far;

    const int b    = blockIdx.x;
    const int tid  = threadIdx.x;
    const int lane = tid & 31;
    const int wv   = tid >> 5;

    const float* base = xyz + (size_t)b * NPTS * 3;

    // Async-stage 48KB of xyz into LDS: 3072 x b128, 12 per thread.
    {
        const unsigned lbase = lds_off_u32(lxyz);
        for (int jc = tid; jc < (NPTS * 3) / 4; jc += 256) {
            const unsigned loff = lbase + (unsigned)jc * 16u;
            const unsigned long long ga =
                (unsigned long long)(const void*)(base + (size_t)jc * 4);
            asm volatile("global_load_async_to_lds_b128 %0, %1, off"
                         :: "v"(loff), "v"(ga) : "memory");
        }
    }
    for (int j = tid; j < NPTS; j += 256) dist[j] = 1e10f;
    wait_asynccnt0();
    __syncthreads();

    int far = 0;
    for (int i = 0; i < NPOINT; ++i) {
        if (tid == 0) {
            fps_idx[b * NPOINT + i] = far;
            new_xyz[((size_t)b * NPOINT + i) * 3 + 0] = lxyz[far * 3 + 0];
            new_xyz[((size_t)b * NPOINT + i) * 3 + 1] = lxyz[far * 3 + 1];
            new_xyz[((size_t)b * NPOINT + i) * 3 + 2] = lxyz[far * 3 + 2];
        }
        const float cx = lxyz[far * 3 + 0];
        const float cy = lxyz[far * 3 + 1];
        const float cz = lxyz[far * 3 + 2];

        float bv = -1.0f;
        int   bi = 0;
        for (int j = tid; j < NPTS; j += 256) {
            float dx = lxyz[j * 3 + 0] - cx;
            float dy = lxyz[j * 3 + 1] - cy;
            float dz = lxyz[j * 3 + 2] - cz;
            float d  = dx * dx + dy * dy + dz * dz;
            float nd = fminf(dist[j], d);
            dist[j]  = nd;
            if (nd > bv) { bv = nd; bi = j; }   // strict > keeps lowest j in subset
        }
        // wave32 argmax reduction, tie -> lower index
        for (int off = 16; off > 0; off >>= 1) {
            float ov = __shfl_xor(bv, off, 32);
            int   oi = __shfl_xor(bi, off, 32);
            if (ov > bv || (ov == bv && oi < bi)) { bv = ov; bi = oi; }
        }
        if (lane == 0) { rval[wv] = bv; ridx[wv] = bi; }
        __syncthreads();
        if (tid == 0) {
            float v = rval[0]; int ix = ridx[0];
            for (int w = 1; w < 8; ++w) {
                if (rval[w] > v || (rval[w] == v && ridx[w] < ix)) { v = rval[w]; ix = ridx[w]; }
            }
            s_far = ix;
        }
        __syncthreads();
        far = s_far;
    }
}

// ---------------------------------------------------------------------------
// Kernel 2: per (batch, 16-query tile):
//   stage:   batch xyz (48KB) -> LDS via Tensor Data Mover (one D#, wave 0)
//   phase A: D tile (16 x 4096) via V_WMMA_F32_16X16X4_F32 (K=3 padded to 4),
//            dist = (-2*dot + |q|^2) + |p|^2 -> 256KB LDS tile
//   phase B: per query, 32 x wave-argmin-with-exclusion (top-32 ascending,
//            ties to lower index — matches lax.top_k(-dists))
//   phase C: gather neighbors, subtract center, write [diff | center] rows.
// ---------------------------------------------------------------------------
__global__ __launch_bounds__(256) void group_kernel(const float* __restrict__ xyz,
                                                    const float* __restrict__ points,
                                                    const float* __restrict__ new_xyz,
                                                    const int* __restrict__ fps_idx,
                                                    float* __restrict__ out) {
    extern __shared__ float sm[];
    float* dtile  = sm;                         // 16 * 4096 floats
    float* lxyz   = sm + 16 * NPTS;             // 3 * 4096 floats, AoS
    float* qx     = lxyz + 3 * NPTS;            // 16
    float* qy     = qx + 16;
    float* qz     = qy + 16;
    float* qn     = qz + 16;
    float* cbuf   = qn + 16;                    // 16 * 64 floats (centers)
    int*   idxbuf = (int*)(cbuf + 16 * CH);     // 16 * 32 ints

    const int tile = blockIdx.x;                // 0..63
    const int b    = blockIdx.y;                // 0..7
    const int tid  = threadIdx.x;
    const int lane = tid & 31;
    const int wv   = tid >> 5;
    const int sub  = lane & 15;
    const int half = lane >> 4;

    const float* xb = xyz + (size_t)b * NPTS * 3;

    // ---- TDM: stage batch xyz (12288 f32) into LDS with one descriptor ----
    if (wv == 0) {
        const unsigned long long ga = (unsigned long long)(const void*)xb;
        u32x4 g0;
        g0[0] = 1u;                                    // count=1 (valid D#)
        g0[1] = lds_off_u32(lxyz);                     // lds_addr (bytes)
        g0[2] = (unsigned)ga;                          // global_addr[31:0]
        g0[3] = (unsigned)((ga >> 32) & 0x1FFFFFFull)  // global_addr[56:32]
              | 0x80000000u;                           // type=2 ("image")
        i32x8 g1;
        g1[0] = 0x00020000;                            // data_size=2 (4 bytes)
        g1[1] = (int)((unsigned)(NPTS * 3) << 16);     // tensor_dim0[15:0] @ [31:16]
        g1[2] = (1 << 16);                             // tensor_dim0 hi=0 | tensor_dim1=1
        g1[3] = (int)((unsigned)(NPTS * 3) << 16);     // tensor_dim1 hi=0 | tile_dim0=12288
        g1[4] = 1;                                     // tile_dim1=1 | tile_dim2=0
        g1[5] = NPTS * 3;                              // tensor_dim0_stride[31:0]
        g1[6] = 0;                                     // stride0 hi | stride1 lo
        g1[7] = 0;
        const i32x4 z4 = {0, 0, 0, 0};
        const i32x8 z8 = {0, 0, 0, 0, 0, 0, 0, 0};
        // 6-arg form (clang-23 / therock lane): groups 2/3 + extended group
        // unused for a <=2-D tensor; cpol=0.
        __builtin_amdgcn_tensor_load_to_lds(g0, g1, z4, z4, z8, 0);
    }

    if (tid < 16) {
        int q = tile * 16 + tid;
        float x = new_xyz[((size_t)b * NPOINT + q) * 3 + 0];
        float y = new_xyz[((size_t)b * NPOINT + q) * 3 + 1];
        float z = new_xyz[((size_t)b * NPOINT + q) * 3 + 2];
        qx[tid] = x; qy[tid] = y; qz[tid] = z;
        qn[tid] = x * x + y * y + z * z;
    }
    __builtin_amdgcn_s_wait_tensorcnt(0);   // no-op for waves with TENSORcnt==0
    __syncthreads();

    // A fragment: 16x4 f32. Lane sub = row M. VGPR0 holds K = 2*half, VGPR1 K = 2*half+1.
    v2f afrag;
    afrag.x = half ? qz[sub] : qx[sub];
    afrag.y = half ? 0.0f    : qy[sub];

    // Loop-invariant query norms for this lane's 8 output rows (registers, not LDS).
    float qreg[8];
#pragma unroll
    for (int r = 0; r < 8; ++r) qreg[r] = qn[r + 8 * half];

    for (int c = wv; c < NPTS / 16; c += 8) {   // uniform per wave -> EXEC all ones
        const int p = c * 16 + sub;
        const float px = lxyz[p * 3 + 0];
        const float py = lxyz[p * 3 + 1];
        const float pz = lxyz[p * 3 + 2];
        const float pn = px * px + py * py + pz * pz;

        // B fragment: 4x16 f32 (KxN). Lane sub = col N. VGPR0 row K=2*half, VGPR1 K=2*half+1.
        v2f bfrag;
        bfrag.x = half ? pz   : px;
        bfrag.y = half ? 0.0f : py;

        v8f acc = {};
        acc = __builtin_amdgcn_wmma_f32_16x16x4_f32(
            /*neg_a=*/false, afrag, /*neg_b=*/false, bfrag,
            /*c_mod=*/(short)0, acc, /*reuse_a=*/false, /*reuse_b=*/false);

#pragma unroll
        for (int r = 0; r < 8; ++r) {
            const int m = r + 8 * half;           // D layout: VGPR r -> M = r + 8*half, N = sub
            dtile[m * NPTS + p] = (-2.0f * acc[r] + qreg[r]) + pn;
        }
    }

    // stage the 16 center rows (new_points) into LDS: 16 x 64 floats
    {
        const int q  = tid >> 4;            // 0..15
        const int c0 = (tid & 15) * 4;      // 0..60
        const int ci = fps_idx[b * NPOINT + tile * 16 + q];
        const float* prow = points + ((size_t)b * NPTS + ci) * CH;
        float4 v = *(const float4*)(prow + c0);
        *(float4*)(cbuf + q * CH + c0) = v;
    }
    __syncthreads();

    // selection: wave wv handles queries wv and wv+8
    for (int qq = wv; qq < 16; qq += 8) {
        float* row = dtile + qq * NPTS;
        for (int it = 0; it < KNBR; ++it) {
            float bv = 3.4e38f;
            int   bi = 0x7fffffff;
            for (int t = 0; t < NPTS / 32; ++t) {
                const int j = t * 32 + lane;       // strided: conflict-free banks
                const float v = row[j];
                if (v < bv) { bv = v; bi = j; }    // strict < keeps lowest j in lane
            }
            for (int off = 16; off > 0; off >>= 1) {
                float ov = __shfl_xor(bv, off, 32);
                int   oi = __shfl_xor(bi, off, 32);
                if (ov < bv || (ov == bv && oi < bi)) { bv = ov; bi = oi; }
            }
            if (lane == 0) {
                idxbuf[qq * KNBR + it] = bi;
                row[bi] = __builtin_inff();        // exclude for next iteration
            }
            __builtin_amdgcn_wave_barrier();
        }
    }
    __syncthreads();

    // gather + normalize + concat: out[b][q][k][0:64]=pt-ctr, [64:128]=ctr
    const float* pbase = points + (size_t)b * NPTS * CH;
    float* obase = out + (((size_t)b * NPOINT + (size_t)tile * 16) * KNBR) * (2 * CH);
    for (int pr = tid; pr < 16 * KNBR; pr += 256) {
        const int q  = pr >> 5;
        const int j  = pr & 31;
        const int id = idxbuf[q * KNBR + j];
        const float4* gp = (const float4*)(pbase + (size_t)id * CH);
        const float4* cp = (const float4*)(cbuf + q * CH);
        float4* op = (float4*)(obase + ((size_t)q * KNBR + j) * (2 * CH));
#pragma unroll
        for (int c4 = 0; c4 < CH / 4; ++c4) {
            float4 g  = gp[c4];
            float4 cv = cp[c4];
            float4 d  = make_float4(g.x - cv.x, g.y - cv.y, g.z - cv.z, g.w - cv.w);
            op[c4]          = d;   // grouped_norm
            op[CH / 4 + c4] = cv;  // broadcast center
        }
    }
}

// ---------------------------------------------------------------------------
extern "C" void kernel_launch(void* const* d_in, const int* in_sizes, int n_in,
                              void* d_out, int out_size, void* d_ws, size_t ws_size,
                              hipStream_t stream) {
    (void)in_sizes; (void)n_in; (void)out_size; (void)ws_size;
    const float* xyz    = (const float*)d_in[0];   // (8, 4096, 3)
    const float* points = (const float*)d_in[1];   // (8, 4096, 64)

    float* new_xyz = (float*)d_out;                        // 8*1024*3 floats
    float* outp    = new_xyz + (size_t)BATCH * NPOINT * 3; // 8*1024*32*128 floats
    int*   fpsidx  = (int*)d_ws;                           // 8*1024 ints

    // Kernel 1: FPS (LDS: 12288 xyz + 4096 dist floats = 64KB dynamic)
    fps_kernel<<<dim3(BATCH), 256, 4 * NPTS * sizeof(float), stream>>>(
        xyz, new_xyz, fpsidx);

    // Kernel 2: TDM-staged xyz + dist tile + top-k + group
    const size_t sh2 = (size_t)(16 * NPTS + 3 * NPTS + 64 + 16 * CH) * sizeof(float)
                     + (size_t)(16 * KNBR) * sizeof(int);
    group_kernel<<<dim3(NPOINT / 16, BATCH), 256, sh2, stream>>>(
        xyz, points, new_xyz, fpsidx, outp);
}